// GATModel_2645699854675
// MI455X (gfx1250) — compile-verified
//
#include <hip/hip_runtime.h>
#include <math.h>

// ---------------------------------------------------------------------------
// GATv2 (2 layers, H=4, D=128) + sum pool + linear, for gfx1250 (MI455X).
// Main matmul uses V_WMMA_F32_16X16X4_F32 (native f32 WMMA, wave32).
// ---------------------------------------------------------------------------

typedef __attribute__((ext_vector_type(2))) float v2f;
typedef __attribute__((ext_vector_type(8))) float v8f;

#define HEADS 4
#define DIM   128
#define HD    512   // HEADS*DIM

// ---------------------------------------------------------------------------
// Generic fills
// ---------------------------------------------------------------------------
__global__ void fill_kernel(float* __restrict__ p, float v, long n) {
    long t = (long)blockIdx.x * blockDim.x + threadIdx.x;
    if (t < n) p[t] = v;
}

__global__ void fill4_kernel(float4* __restrict__ p, float v, long n4) {
    long t = (long)blockIdx.x * blockDim.x + threadIdx.x;
    if (t < n4) p[t] = make_float4(v, v, v, v);
}

// ---------------------------------------------------------------------------
// C[M,512] = A[M,128] @ W[128,512] + bias[512]
// One wave (32 threads) per 16-row x 64-col slab.
// grid = (ceil(M/16), 512/64=8), block = 32.
// A-fragment (16x4 f32): lane<16 holds (M=lane, K=kb+0..1); lane>=16 holds
// (M=lane-16, K=kb+2..3). B-fragment (4x16) mirrors this. C/D: VGPR i holds
// row i (lanes 0-15) and row i+8 (lanes 16-31), col = lane&15.
// Partial last row-tile: A row index is clamped (a row of A only feeds the
// same row of C, and out-of-range C rows are never stored), so loads are
// unconditional -> no exec-mask churn in the K-loop.
// ---------------------------------------------------------------------------
__global__ __launch_bounds__(32) void gemm_wmma_kernel(
    const float* __restrict__ A, const float* __restrict__ W,
    const float* __restrict__ bias, float* __restrict__ C, int M) {
    const int lane = threadIdx.x & 31;
    const int half = lane >> 4;        // 0 or 1
    const int l15  = lane & 15;
    const int rowbase = blockIdx.x * 16;
    const int row  = rowbase + l15;
    const int rowc = row < (M - 1) ? row : (M - 1);   // clamped, always valid
    const int col0 = blockIdx.y * 64;

    v8f acc[4];
    #pragma unroll
    for (int t = 0; t < 4; ++t) {
        float bv = bias[col0 + t * 16 + l15];
        #pragma unroll
        for (int i = 0; i < 8; ++i) acc[t][i] = bv;
    }

    // A row as v2f: element j holds floats {2j, 2j+1}; k = kb + half*2 is even.
    const v2f* arow2 = (const v2f*)(A + (size_t)rowc * DIM);

    for (int kb = 0; kb < DIM; kb += 4) {
        const int k = kb + half * 2;
        v2f a = arow2[k >> 1];                       // one b64 load
        #pragma unroll
        for (int t = 0; t < 4; ++t) {
            const int c = col0 + t * 16 + l15;
            v2f b;
            b.x = W[(size_t)k * HD + c];
            b.y = W[(size_t)(k + 1) * HD + c];
            acc[t] = __builtin_amdgcn_wmma_f32_16x16x4_f32(
                false, a, false, b, (short)0, acc[t], false, false);
        }
    }

    if (rowbase + 16 <= M) {
        // full tile: unconditional stores
        #pragma unroll
        for (int t = 0; t < 4; ++t) {
            const int c = col0 + t * 16 + l15;
            #pragma unroll
            for (int i = 0; i < 8; ++i) {
                const int r = rowbase + i + half * 8;
                C[(size_t)r * HD + c] = acc[t][i];
            }
        }
    } else {
        #pragma unroll
        for (int t = 0; t < 4; ++t) {
            const int c = col0 + t * 16 + l15;
            #pragma unroll
            for (int i = 0; i < 8; ++i) {
                const int r = rowbase + i + half * 8;
                if (r < M) C[(size_t)r * HD + c] = acc[t][i];
            }
        }
    }
}

// ---------------------------------------------------------------------------
// h[n, :] = M1[feats[n], :]   (512 floats per row, float4 vectorized)
// thread t -> node n = t>>7, float4 slot j = t&127
// ---------------------------------------------------------------------------
__global__ void gather_rows_kernel(const float* __restrict__ M1,
                                   const int* __restrict__ feats,
                                   float* __restrict__ h, int N) {
    long t = (long)blockIdx.x * blockDim.x + threadIdx.x;
    long total = (long)N * 128;   // N*512 floats / 4
    if (t >= total) return;
    int n = (int)(t >> 7);
    int j = (int)(t & 127);
    const float4* srcp = (const float4*)(M1 + (size_t)feats[n] * HD);
    float4* dstp = (float4*)(h + (size_t)n * HD);
    dstp[j] = srcp[j];
}

// ---------------------------------------------------------------------------
// float atomic max via CAS loop (portable)
// ---------------------------------------------------------------------------
__device__ inline void atomicMaxF(float* addr, float val) {
    unsigned int* ua = (unsigned int*)addr;
    unsigned int old = *ua;
    while (__uint_as_float(old) < val) {
        unsigned int assumed = old;
        old = atomicCAS(ua, assumed, __float_as_uint(val));
        if (old == assumed) break;
    }
}

__device__ inline float lrelu(float v) { return v > 0.0f ? v : 0.2f * v; }

// ---------------------------------------------------------------------------
// Per-edge logits: logit[e,h] = sum_d lrelu(hs[d]+hd[d]) * a[h,d]
// One wave per edge; lane covers 4 contiguous floats per head.
// Also scatter-max into maxb[dst,h].
// ---------------------------------------------------------------------------
__global__ void edge_logits_kernel(const float* __restrict__ h,
                                   const int* __restrict__ src,
                                   const int* __restrict__ dst,
                                   const float* __restrict__ attn,
                                   float* __restrict__ logits,
                                   float* __restrict__ maxb, int E) {
    int wid  = (int)((long)blockIdx.x * (blockDim.x >> 5) + (threadIdx.x >> 5));
    int lane = threadIdx.x & 31;
    if (wid >= E) return;
    const int s = src[wid], d = dst[wid];
    const float4* hs = (const float4*)(h + (size_t)s * HD);
    const float4* hd = (const float4*)(h + (size_t)d * HD);
    const float4* av = (const float4*)attn;
    #pragma unroll
    for (int hh = 0; hh < HEADS; ++hh) {
        float4 vs = hs[hh * 32 + lane];
        float4 vd = hd[hh * 32 + lane];
        float4 va = av[hh * 32 + lane];
        float sum = lrelu(vs.x + vd.x) * va.x
                  + lrelu(vs.y + vd.y) * va.y
                  + lrelu(vs.z + vd.z) * va.z
                  + lrelu(vs.w + vd.w) * va.w;
        #pragma unroll
        for (int off = 16; off > 0; off >>= 1)
            sum += __shfl_xor(sum, off, 32);
        if (lane == 0) {
            logits[(size_t)wid * HEADS + hh] = sum;
            atomicMaxF(&maxb[(size_t)d * HEADS + hh], sum);
        }
    }
}

// ---------------------------------------------------------------------------
// ex = exp(logit - max[dst]); accumulate denominators. Thread per (e,h).
// ---------------------------------------------------------------------------
__global__ void edge_exp_kernel(float* __restrict__ logits,
                                const int* __restrict__ dst,
                                const float* __restrict__ maxb,
                                float* __restrict__ den, int E) {
    long t = (long)blockIdx.x * blockDim.x + threadIdx.x;
    if (t >= (long)E * HEADS) return;
    int e  = (int)(t >> 2);
    int hh = (int)(t & 3);
    int d  = dst[e];
    float ex = __expf(logits[t] - maxb[(size_t)d * HEADS + hh]);
    logits[t] = ex;
    atomicAdd(&den[(size_t)d * HEADS + hh], ex);
}

// ---------------------------------------------------------------------------
// rst[dst] += alpha * h[src].  One wave per edge.
// ---------------------------------------------------------------------------
__global__ void edge_agg_kernel(const float* __restrict__ h,
                                const int* __restrict__ src,
                                const int* __restrict__ dst,
                                const float* __restrict__ logits,
                                const float* __restrict__ den,
                                float* __restrict__ rst, int E) {
    int wid  = (int)((long)blockIdx.x * (blockDim.x >> 5) + (threadIdx.x >> 5));
    int lane = threadIdx.x & 31;
    if (wid >= E) return;
    const int s = src[wid], d = dst[wid];
    const float4* hs = (const float4*)(h + (size_t)s * HD);
    float* rd = rst + (size_t)d * HD;
    #pragma unroll
    for (int hh = 0; hh < HEADS; ++hh) {
        float alpha = logits[(size_t)wid * HEADS + hh]
                    / den[(size_t)d * HEADS + hh];
        float4 vs = hs[hh * 32 + lane];
        int base = hh * DIM + lane * 4;
        atomicAdd(&rd[base + 0], alpha * vs.x);
        atomicAdd(&rd[base + 1], alpha * vs.y);
        atomicAdd(&rd[base + 2], alpha * vs.z);
        atomicAdd(&rd[base + 3], alpha * vs.w);
    }
}

// ---------------------------------------------------------------------------
// x[n,d] = mean over heads of rst[n,h,d].  Thread per (n, float4-slot).
// ---------------------------------------------------------------------------
__global__ void head_mean_kernel(const float* __restrict__ rst,
                                 float* __restrict__ x, int N) {
    long t = (long)blockIdx.x * blockDim.x + threadIdx.x;
    if (t >= (long)N * 32) return;   // 128 floats / 4 per node
    int n = (int)(t >> 5);
    int j = (int)(t & 31);
    const float4* r = (const float4*)(rst + (size_t)n * HD);
    float4 a = r[j], b = r[32 + j], c = r[64 + j], d = r[96 + j];
    float4 o;
    o.x = 0.25f * (a.x + b.x + c.x + d.x);
    o.y = 0.25f * (a.y + b.y + c.y + d.y);
    o.z = 0.25f * (a.z + b.z + c.z + d.z);
    o.w = 0.25f * (a.w + b.w + c.w + d.w);
    ((float4*)(x + (size_t)n * DIM))[j] = o;
}

// ---------------------------------------------------------------------------
// out[g] += dot(x[n,:], Wreg)  fused sum-pool + regression. One wave per node.
// ---------------------------------------------------------------------------
__global__ void pool_kernel(const float* __restrict__ x,
                            const int* __restrict__ gid,
                            const float* __restrict__ wreg,
                            float* __restrict__ out, int N) {
    int wid  = (int)((long)blockIdx.x * (blockDim.x >> 5) + (threadIdx.x >> 5));
    int lane = threadIdx.x & 31;
    if (wid >= N) return;
    float4 vx = ((const float4*)(x + (size_t)wid * DIM))[lane];
    float4 vw = ((const float4*)wreg)[lane];
    float sum = vx.x * vw.x + vx.y * vw.y + vx.z * vw.z + vx.w * vw.w;
    #pragma unroll
    for (int off = 16; off > 0; off >>= 1)
        sum += __shfl_xor(sum, off, 32);
    if (lane == 0) atomicAdd(&out[gid[wid]], sum);
}

// ---------------------------------------------------------------------------
// Host-side launcher
// ---------------------------------------------------------------------------
extern "C" void kernel_launch(void* const* d_in, const int* in_sizes, int n_in,
                              void* d_out, int out_size, void* d_ws, size_t ws_size,
                              hipStream_t stream) {
    const int*   feats = (const int*)  d_in[0];
    const int*   src   = (const int*)  d_in[1];
    const int*   dst   = (const int*)  d_in[2];
    const int*   gid   = (const int*)  d_in[3];
    const float* emb   = (const float*)d_in[4];
    const float* W1    = (const float*)d_in[5];
    const float* b1    = (const float*)d_in[6];
    const float* a1    = (const float*)d_in[7];
    const float* W2    = (const float*)d_in[8];
    const float* b2    = (const float*)d_in[9];
    const float* a2    = (const float*)d_in[10];
    const float* Wreg  = (const float*)d_in[11];
    float* out = (float*)d_out;

    const int N = in_sizes[0];
    const int E = in_sizes[1];
    const int V = in_sizes[4] / DIM;   // 128
    const int G = out_size;            // 64

    // workspace carve-out
    char* ws = (char*)d_ws;
    size_t off = 0;
    auto carve = [&](size_t bytes) -> void* {
        void* p = ws + off;
        off = (off + bytes + 255) & ~(size_t)255;
        return p;
    };
    float* h    = (float*)carve((size_t)N * HD * sizeof(float));     // [N,4,128]
    float* rst  = (float*)carve((size_t)N * HD * sizeof(float));     // [N,4,128]
    float* logi = (float*)carve((size_t)E * HEADS * sizeof(float));  // [E,4]
    float* maxb = (float*)carve((size_t)N * HEADS * sizeof(float));  // [N,4]
    float* den  = (float*)carve((size_t)N * HEADS * sizeof(float));  // [N,4]
    float* x    = (float*)carve((size_t)N * DIM * sizeof(float));    // [N,128]
    float* M1   = (float*)carve((size_t)V * HD * sizeof(float));     // [128,512]
    (void)ws_size; (void)n_in;

    const int blk = 256;
    const int wavesPerBlk = blk / 32;
    const long NHD4 = (long)N * HD / 4;   // float4 count for rst
    const long NH   = (long)N * HEADS;
    const long EH   = (long)E * HEADS;
    const int  egrd = (E + wavesPerBlk - 1) / wavesPerBlk;

    // ---- Layer 1: h1 = (emb @ W1 + b1)[feats] -----------------------------
    gemm_wmma_kernel<<<dim3((V + 15) / 16, HD / 64), 32, 0, stream>>>(emb, W1, b1, M1, V);
    gather_rows_kernel<<<(int)(((long)N * 128 + blk - 1) / blk), blk, 0, stream>>>(M1, feats, h, N);

    for (int layer = 0; layer < 2; ++layer) {
        const float* attn = (layer == 0) ? a1 : a2;

        fill_kernel<<<(int)((NH + blk - 1) / blk), blk, 0, stream>>>(maxb, -INFINITY, NH);
        fill_kernel<<<(int)((NH + blk - 1) / blk), blk, 0, stream>>>(den, 0.0f, NH);
        fill4_kernel<<<(int)((NHD4 + blk - 1) / blk), blk, 0, stream>>>((float4*)rst, 0.0f, NHD4);

        edge_logits_kernel<<<egrd, blk, 0, stream>>>(h, src, dst, attn, logi, maxb, E);
        edge_exp_kernel<<<(int)((EH + blk - 1) / blk), blk, 0, stream>>>(logi, dst, maxb, den, E);
        edge_agg_kernel<<<egrd, blk, 0, stream>>>(h, src, dst, logi, den, rst, E);
        head_mean_kernel<<<(int)(((long)N * 32 + blk - 1) / blk), blk, 0, stream>>>(rst, x, N);

        if (layer == 0) {
            // h2 = x1 @ W2 + b2  (full WMMA GEMM over 50k rows)
            gemm_wmma_kernel<<<dim3((N + 15) / 16, HD / 64), 32, 0, stream>>>(x, W2, b2, h, N);
        }
    }

    // ---- Pool + regression ------------------------------------------------
    fill_kernel<<<1, 256, 0, stream>>>(out, 0.0f, G);
    pool_kernel<<<(N + wavesPerBlk - 1) / wavesPerBlk, blk, 0, stream>>>(x, gid, Wreg, out, N);
}